// GraphRegressor_66838281061050
// MI455X (gfx1250) — compile-verified
//
#include <hip/hip_runtime.h>
#include <hip/hip_bf16.h>

typedef __attribute__((ext_vector_type(2))) float v2f;
typedef __attribute__((ext_vector_type(8))) float v8f;

#define N_GRAPHS 64
#define POOL_F   24

// ---------------- utility kernels ----------------

__global__ void zero_f32(float* __restrict__ p, long long n) {
    long long i = (long long)blockIdx.x * blockDim.x + threadIdx.x;
    if (i < n) p[i] = 0.0f;
}

__global__ void deg_kernel(const int* __restrict__ dst, float* __restrict__ deg, int E) {
    int e = blockIdx.x * blockDim.x + threadIdx.x;
    // speculative prefetch ahead on the edge list (gfx1250 global_prefetch_b8)
    __builtin_prefetch(dst + e + 8192, 0, 0);
    if (e < E) atomicAdd(&deg[dst[e]], 1.0f);
}

__global__ void dinv_kernel(const float* __restrict__ deg, float* __restrict__ dinv, int N) {
    int n = blockIdx.x * blockDim.x + threadIdx.x;
    if (n < N) {
        float d = deg[n] + 1.0f;   // +1 self-loop
        dinv[n] = rsqrtf(d);
    }
}

// ---------------- WMMA GEMM: T = [relu](H) @ W ; OutInit = bias + dinv^2 * T ----------------
// One 16-row tile per wave, 8 waves per 256-thread block.
// Uses V_WMMA_F32_16X16X4_F32 (fp32 in / fp32 acc), K stepped by 4.

template<int IN, int IN_P, int OUT, int OUT_P, bool RELU_IN>
__global__ __launch_bounds__(256)
void gcn_gemm(const float* __restrict__ H, const float* __restrict__ W,
              const float* __restrict__ bias, const float* __restrict__ dinv,
              float* __restrict__ T, float* __restrict__ OutInit, int N) {
    __shared__ float sW[IN_P * OUT_P];
    for (int i = threadIdx.x; i < IN_P * OUT_P; i += blockDim.x) {
        int kr = i / OUT_P, kc = i % OUT_P;
        sW[i] = (kr < IN && kc < OUT) ? W[kr * OUT + kc] : 0.0f;   // zero-pad
    }
    __syncthreads();

    const int lane  = threadIdx.x & 31;
    const int wave  = threadIdx.x >> 5;
    const int m     = lane & 15;          // col within 16-wide tile / row-of-A index
    const int khalf = (lane >> 4) * 2;    // lanes 16..31 hold K=2,3 of each K-step
    const int rowBase = (blockIdx.x * 8 + wave) * 16;
    if (rowBase >= N) return;             // uniform per wave -> EXEC stays all-ones

    int r = rowBase + m;
    if (r >= N) r = N - 1;                // clamp loads; stores are masked below

    v8f acc[OUT_P / 16] = {};

    for (int k0 = 0; k0 < IN_P; k0 += 4) {
        const int ka = k0 + khalf;
        float a0 = (ka     < IN) ? H[(long long)r * IN + ka]     : 0.0f;
        float a1 = (ka + 1 < IN) ? H[(long long)r * IN + ka + 1] : 0.0f;
        if (RELU_IN) { a0 = fmaxf(a0, 0.0f); a1 = fmaxf(a1, 0.0f); }
        v2f a; a.x = a0; a.y = a1;
#pragma unroll
        for (int j = 0; j < OUT_P / 16; ++j) {
            v2f b;
            b.x = sW[ka       * OUT_P + j * 16 + m];
            b.y = sW[(ka + 1) * OUT_P + j * 16 + m];
            acc[j] = __builtin_amdgcn_wmma_f32_16x16x4_f32(
                false, a, false, b, (short)0, acc[j], false, false);
        }
    }

    // C/D layout: VGPR v -> M=v (lanes 0-15) / M=v+8 (lanes 16-31), N = lane&15
    const int rowOff = (lane >> 4) * 8;

    // hoist bias per column tile (compile-time col mask when OUT == OUT_P)
    float bcol[OUT_P / 16];
#pragma unroll
    for (int j = 0; j < OUT_P / 16; ++j)
        bcol[j] = (OUT == OUT_P || j * 16 + m < OUT) ? bias[j * 16 + m] : 0.0f;

    if (rowBase + 16 <= N) {
        // fast path: full 16-row tile, no row guards, no exec churn
#pragma unroll
        for (int v = 0; v < 8; ++v) {
            const int row = rowBase + rowOff + v;
            const float di = dinv[row];
            const float di2 = di * di;
#pragma unroll
            for (int j = 0; j < OUT_P / 16; ++j) {
                const int col = j * 16 + m;
                if (OUT == OUT_P || col < OUT) {
                    const long long idx = (long long)row * OUT + col;
                    const float val = acc[j][v];
                    T[idx] = val;
                    OutInit[idx] = bcol[j] + di2 * val;
                }
            }
        }
    } else {
        // tail tile: per-row guard
#pragma unroll
        for (int v = 0; v < 8; ++v) {
            const int row = rowBase + rowOff + v;
            if (row < N) {
                const float di = dinv[row];
                const float di2 = di * di;
#pragma unroll
                for (int j = 0; j < OUT_P / 16; ++j) {
                    const int col = j * 16 + m;
                    if (OUT == OUT_P || col < OUT) {
                        const long long idx = (long long)row * OUT + col;
                        const float val = acc[j][v];
                        T[idx] = val;
                        OutInit[idx] = bcol[j] + di2 * val;
                    }
                }
            }
        }
    }
}

// ---------------- edge scatter: Out[dst] += dinv[s]*dinv[d] * T[src] ----------------
// FPE threads per edge (power of two), 4 features per thread:
// b128 gathers from T, fp32 global atomics into Out.

template<int F, int FPE>
__global__ __launch_bounds__(256)
void scatter_edges_v4(const int* __restrict__ src, const int* __restrict__ dst,
                      const float* __restrict__ dinv, const float* __restrict__ T,
                      float* __restrict__ Out, int E) {
    long long gid = (long long)blockIdx.x * blockDim.x + threadIdx.x;
    long long e = gid / FPE;               // FPE is a power of two -> shift
    int f = (int)(gid & (FPE - 1)) * 4;
    if (e < E && f < F) {
        int s = src[e], d = dst[e];
        float w = dinv[s] * dinv[d];
        const float4 t = *(const float4*)(T + (long long)s * F + f);  // global_load_b128
        float* o = Out + (long long)d * F + f;
        atomicAdd(o + 0, w * t.x);
        atomicAdd(o + 1, w * t.y);
        atomicAdd(o + 2, w * t.z);
        atomicAdd(o + 3, w * t.w);
    }
}

// ---------------- pooling ----------------

__global__ __launch_bounds__(256)
void pool_kernel(const float* __restrict__ Y, const int* __restrict__ batch,
                 float* __restrict__ pooled, float* __restrict__ counts, int N) {
    __shared__ float sPool[N_GRAPHS * POOL_F];
    __shared__ float sCnt[N_GRAPHS];
    for (int i = threadIdx.x; i < N_GRAPHS * POOL_F; i += blockDim.x) sPool[i] = 0.0f;
    for (int i = threadIdx.x; i < N_GRAPHS; i += blockDim.x) sCnt[i] = 0.0f;
    __syncthreads();

    int n = blockIdx.x * blockDim.x + threadIdx.x;
    if (n < N) {
        int g = batch[n];
        atomicAdd(&sCnt[g], 1.0f);
        for (int f = 0; f < POOL_F; ++f)
            atomicAdd(&sPool[g * POOL_F + f], Y[(long long)n * POOL_F + f]);
    }
    __syncthreads();

    // batch is sorted -> most partials are zero; adding 0 is a numeric no-op, skip it
    for (int i = threadIdx.x; i < N_GRAPHS * POOL_F; i += blockDim.x)
        if (sPool[i] != 0.0f) atomicAdd(&pooled[i], sPool[i]);
    for (int i = threadIdx.x; i < N_GRAPHS; i += blockDim.x)
        if (sCnt[i] != 0.0f) atomicAdd(&counts[i], sCnt[i]);
}

__global__ void finalize_kernel(const float* __restrict__ pooled,
                                const float* __restrict__ counts,
                                float* __restrict__ out) {
    int i = blockIdx.x * blockDim.x + threadIdx.x;
    if (i < N_GRAPHS * POOL_F) {
        int g = i / POOL_F;
        out[i] = tanhf(pooled[i] / fmaxf(counts[g], 1.0f));
    }
}

// ---------------- host launch ----------------

extern "C" void kernel_launch(void* const* d_in, const int* in_sizes, int n_in,
                              void* d_out, int out_size, void* d_ws, size_t ws_size,
                              hipStream_t stream) {
    const float* pos   = (const float*)d_in[0];
    const int*   eidx  = (const int*)d_in[1];     // [2, E] (harness integer -> int32)
    const int*   batch = (const int*)d_in[2];
    const float* W1 = (const float*)d_in[3];
    const float* b1 = (const float*)d_in[4];
    const float* W2 = (const float*)d_in[5];
    const float* b2 = (const float*)d_in[6];
    const float* W3 = (const float*)d_in[7];
    const float* b3 = (const float*)d_in[8];

    const int N = in_sizes[0] / 3;
    const int E = in_sizes[1] / 2;
    const int* srcI = eidx;
    const int* dstI = eidx + E;

    // workspace layout (floats)
    float* ws = (float*)d_ws;
    float* deg    = ws;                    // N
    float* dinv   = deg + N;               // N
    float* B1     = dinv + N;              // N*64
    float* B2     = B1 + (long long)N*64;  // N*64
    float* B3     = B2 + (long long)N*64;  // N*64
    float* pooled = B3 + (long long)N*64;  // 64*24
    float* counts = pooled + N_GRAPHS*POOL_F; // 64

    const int TB = 256;
    // zero deg + pooled + counts (fresh every call)
    zero_f32<<<(N + TB - 1) / TB, TB, 0, stream>>>(deg, N);
    zero_f32<<<1, TB, 0, stream>>>(pooled, N_GRAPHS * POOL_F + N_GRAPHS);

    // degree -> dinv
    deg_kernel<<<(E + TB - 1) / TB, TB, 0, stream>>>(dstI, deg, E);
    dinv_kernel<<<(N + TB - 1) / TB, TB, 0, stream>>>(deg, dinv, N);

    const int gemmBlocks = (N + 127) / 128;
    long long t16 = (long long)E * 16, t8 = (long long)E * 8;
    const int sc16 = (int)((t16 + TB - 1) / TB);
    const int sc8  = (int)((t8  + TB - 1) / TB);

    // Layer 1: pos(3) -> 64
    gcn_gemm<3, 4, 64, 64, false><<<gemmBlocks, TB, 0, stream>>>(pos, W1, b1, dinv, B1, B2, N);
    scatter_edges_v4<64, 16><<<sc16, TB, 0, stream>>>(srcI, dstI, dinv, B1, B2, E);

    // Layer 2: 64 -> 64 (ReLU fused into A-load)
    gcn_gemm<64, 64, 64, 64, true><<<gemmBlocks, TB, 0, stream>>>(B2, W2, b2, dinv, B3, B1, N);
    scatter_edges_v4<64, 16><<<sc16, TB, 0, stream>>>(srcI, dstI, dinv, B3, B1, E);

    // Layer 3: 64 -> 24 (ReLU fused into A-load)
    gcn_gemm<64, 64, 24, 32, true><<<gemmBlocks, TB, 0, stream>>>(B1, W3, b3, dinv, B2, B3, N);
    scatter_edges_v4<24, 8><<<sc8, TB, 0, stream>>>(srcI, dstI, dinv, B2, B3, E);

    // Pool + tanh
    pool_kernel<<<(N + TB - 1) / TB, TB, 0, stream>>>(B3, batch, pooled, counts, N);
    finalize_kernel<<<(N_GRAPHS * POOL_F + TB - 1) / TB, TB, 0, stream>>>(pooled, counts, (float*)d_out);
}